// Mamba2_11355893530799
// MI455X (gfx1250) — compile-verified
//
#include <hip/hip_runtime.h>
#include <hip/hip_bf16.h>

// ---------------- problem constants ----------------
#define TOK    4096   // BATCH*SEQ
#define SEQL   2048
#define DIMK   2048   // DIM
#define DIP    8480   // D_IN_PROJ
#define DINNER 4096
#define CONVD  4352   // D_INNER + 2*D_STATE
#define NH     32
#define HD     128    // head dim p
#define DST    128    // d_state n
#define NCH    8      // chunks per sequence (2048/256)
#define CHL    256    // chunk length

typedef float v2f __attribute__((ext_vector_type(2)));
typedef float v4f __attribute__((ext_vector_type(4)));
typedef float v8f __attribute__((ext_vector_type(8)));

// 8 K-steps of 4 over a 128x(32) LDS tile pair; wave owns 32x64 (2x4 WMMA tiles).
// Uses locals: wm, wn, lrow, khalf, acc.
#define WMMA_TILE_STEP(AS, BS)                                                             \
    _Pragma("unroll")                                                                      \
    for (int kk = 0; kk < 32; kk += 4) {                                                   \
        const int ko = kk + (khalf << 1);                                                  \
        v2f a0 = *(const v2f*)&AS[wm +      lrow][ko];                                     \
        v2f a1 = *(const v2f*)&AS[wm + 16 + lrow][ko];                                     \
        v2f b0 = *(const v2f*)&BS[wn +      lrow][ko];                                     \
        v2f b1 = *(const v2f*)&BS[wn + 16 + lrow][ko];                                     \
        v2f b2 = *(const v2f*)&BS[wn + 32 + lrow][ko];                                     \
        v2f b3 = *(const v2f*)&BS[wn + 48 + lrow][ko];                                     \
        acc[0][0] = __builtin_amdgcn_wmma_f32_16x16x4_f32(false, a0, false, b0, (short)0, acc[0][0], false, false); \
        acc[0][1] = __builtin_amdgcn_wmma_f32_16x16x4_f32(false, a0, false, b1, (short)0, acc[0][1], false, false); \
        acc[0][2] = __builtin_amdgcn_wmma_f32_16x16x4_f32(false, a0, false, b2, (short)0, acc[0][2], false, false); \
        acc[0][3] = __builtin_amdgcn_wmma_f32_16x16x4_f32(false, a0, false, b3, (short)0, acc[0][3], false, false); \
        acc[1][0] = __builtin_amdgcn_wmma_f32_16x16x4_f32(false, a1, false, b0, (short)0, acc[1][0], false, false); \
        acc[1][1] = __builtin_amdgcn_wmma_f32_16x16x4_f32(false, a1, false, b1, (short)0, acc[1][1], false, false); \
        acc[1][2] = __builtin_amdgcn_wmma_f32_16x16x4_f32(false, a1, false, b2, (short)0, acc[1][2], false, false); \
        acc[1][3] = __builtin_amdgcn_wmma_f32_16x16x4_f32(false, a1, false, b3, (short)0, acc[1][3], false, false); \
    }

// =====================================================================
// Generic NT GEMM:  C[m,n] = sum_k A[m,k]*B[n,k] (+bias[n]), batched via
// grid.z with element strides.  f32 WMMA 16x16x4, 128x128x32 block tile.
// =====================================================================
__global__ __launch_bounds__(256) void gemm_nt_wmma(
    const float* __restrict__ A, long long sAb, int lda,
    const float* __restrict__ B, long long sBb, int ldb,
    const float* __restrict__ bias,
    float* __restrict__ C, long long sCb, int ldc,
    int M, int N, int K)
{
    A += sAb * (long long)blockIdx.z;
    B += sBb * (long long)blockIdx.z;
    C += sCb * (long long)blockIdx.z;
    const int tileM = blockIdx.y * 128;
    const int tileN = blockIdx.x * 128;

    __shared__ __align__(16) float As[128][34];
    __shared__ __align__(16) float Bs[128][34];

    const int tid   = threadIdx.x;
    const int wave  = tid >> 5;
    const int lane  = tid & 31;
    const int wm    = (wave >> 1) << 5;
    const int wn    = (wave & 1) << 6;
    const int lrow  = lane & 15;
    const int khalf = lane >> 4;

    v8f acc[2][4] = {};

    for (int k0 = 0; k0 < K; k0 += 32) {
        #pragma unroll
        for (int i = 0; i < 4; ++i) {
            int idx = tid + (i << 8);
            int r   = idx >> 3;
            int kq  = (idx & 7) << 2;
            int gm  = tileM + r;
            v4f av = {0.f, 0.f, 0.f, 0.f};
            if (gm < M) av = *(const v4f*)(A + (size_t)gm * lda + k0 + kq);
            As[r][kq + 0] = av[0]; As[r][kq + 1] = av[1];
            As[r][kq + 2] = av[2]; As[r][kq + 3] = av[3];
            int gn  = tileN + r;
            v4f bv = {0.f, 0.f, 0.f, 0.f};
            if (gn < N) bv = *(const v4f*)(B + (size_t)gn * ldb + k0 + kq);
            Bs[r][kq + 0] = bv[0]; Bs[r][kq + 1] = bv[1];
            Bs[r][kq + 2] = bv[2]; Bs[r][kq + 3] = bv[3];
        }
        // prefetch next K tile (global_prefetch_b8 path)
        if (k0 + 32 < K) {
            int r  = tid >> 3;
            int kq = (tid & 7) << 2;
            int gm = tileM + r, gn = tileN + r;
            if (gm < M) __builtin_prefetch(A + (size_t)gm * lda + k0 + 32 + kq, 0, 3);
            if (gn < N) __builtin_prefetch(B + (size_t)gn * ldb + k0 + 32 + kq, 0, 3);
        }
        __syncthreads();
        WMMA_TILE_STEP(As, Bs)
        __syncthreads();
    }

    #pragma unroll
    for (int mi = 0; mi < 2; ++mi)
        #pragma unroll
        for (int ni = 0; ni < 4; ++ni)
            #pragma unroll
            for (int r = 0; r < 8; ++r) {
                int row = tileM + wm + (mi << 4) + (khalf << 3) + r;
                int col = tileN + wn + (ni << 4) + lrow;
                if (row < M && col < N) {
                    float v = acc[mi][ni][r];
                    if (bias) v += bias[col];
                    C[(size_t)row * ldc + col] = v;
                }
            }
}

// =====================================================================
// conv1d (depthwise, window 4, causal) + SiLU on xBC, softplus on dt
// =====================================================================
__global__ __launch_bounds__(256) void conv_silu_dt(
    const float* __restrict__ zx, const float* __restrict__ convW,
    const float* __restrict__ convB, const float* __restrict__ dtBias,
    float* __restrict__ xbc, float* __restrict__ dtb)
{
    const int t = blockIdx.x;
    const int s = t & (SEQL - 1);
    for (int c = threadIdx.x; c < CONVD; c += 256) {
        float acc = convB[c];
        #pragma unroll
        for (int k = 0; k < 4; ++k) {
            int ss = s - 3 + k;
            if (ss >= 0)
                acc += zx[(size_t)(t - 3 + k) * DIP + DINNER + c] * convW[c * 4 + k];
        }
        xbc[(size_t)t * CONVD + c] = acc / (1.f + __expf(-acc));  // SiLU
    }
    if (threadIdx.x < NH) {
        int h = threadIdx.x;
        float v = zx[(size_t)t * DIP + DINNER + CONVD + h] + dtBias[h];
        dtb[t * NH + h] = (v > 20.f) ? v : log1pf(expf(v));       // softplus
    }
}

// =====================================================================
// Acum[b,h,c,l] = cumsum_l( -exp(A_log[h]) * dt[b, c*256+l, h] )
// =====================================================================
__global__ __launch_bounds__(256) void cumsum_kernel(
    const float* __restrict__ dtb, const float* __restrict__ Alog,
    float* __restrict__ acum)
{
    const int id = blockIdx.x;
    const int b = id >> 8, rem = id & 255, h = rem >> 3, c = rem & 7;
    const int l = threadIdx.x;
    const float Ah = -expf(Alog[h]);
    __shared__ float sb[CHL];
    sb[l] = Ah * dtb[((b * SEQL) + (c * CHL + l)) * NH + h];
    __syncthreads();
    for (int off = 1; off < CHL; off <<= 1) {
        float tv = (l >= off) ? sb[l - off] : 0.f;
        __syncthreads();
        sb[l] += tv;
        __syncthreads();
    }
    acum[(((b * NH) + h) * NCH + c) * CHL + l] = sb[l];
}

// =====================================================================
// local chunk states via WMMA:  S[p,n] = sum_l (x[l,p]*w[l]) * B[l,n]
// w[l] = exp(Acum_last - Acum[l]) * dt[l].  One block per (b,c,h).
// M=128 (p), N=128 (n), K=256 (l); both operands transposed during staging.
// =====================================================================
__global__ __launch_bounds__(256) void states_wmma(
    const float* __restrict__ xbc, const float* __restrict__ dtb,
    const float* __restrict__ acum, float* __restrict__ Sl)
{
    const int id = blockIdx.x;
    const int b = id >> 8, rem = id & 255, c = rem >> 5, h = rem & 31;
    const size_t t0 = (size_t)b * SEQL + c * CHL;
    const int hbase = h * HD;

    __shared__ __align__(16) float As[128][34];
    __shared__ __align__(16) float Bs[128][34];
    __shared__ float sW[CHL];

    const int tid   = threadIdx.x;
    const int wave  = tid >> 5;
    const int lane  = tid & 31;
    const int wm    = (wave >> 1) << 5;
    const int wn    = (wave & 1) << 6;
    const int lrow  = lane & 15;
    const int khalf = lane >> 4;

    const float* ac = acum + (((b * NH) + h) * NCH + c) * CHL;
    const float alast = ac[CHL - 1];
    sW[tid] = __expf(alast - ac[tid]) * dtb[(t0 + tid) * NH + h];
    __syncthreads();

    v8f acc[2][4] = {};

    for (int k0 = 0; k0 < CHL; k0 += 32) {
        #pragma unroll
        for (int i = 0; i < 4; ++i) {
            int idx = tid + (i << 8);
            int p   = idx & 127;           // channel (p for A, n for B)
            int l4  = ((idx >> 7) & 7) << 2;
            #pragma unroll
            for (int q = 0; q < 4; ++q) {
                int ll = l4 + q;
                size_t row = (t0 + k0 + ll) * (size_t)CONVD;
                As[p][ll] = xbc[row + hbase + p] * sW[k0 + ll];   // Xd^T, decay folded
                Bs[p][ll] = xbc[row + DINNER + p];                // Bc^T
            }
        }
        __syncthreads();
        WMMA_TILE_STEP(As, Bs)
        __syncthreads();
    }

    size_t base = (size_t)(((b * NCH) + c) * NH + h) * (HD * DST);
    #pragma unroll
    for (int mi = 0; mi < 2; ++mi)
        #pragma unroll
        for (int ni = 0; ni < 4; ++ni)
            #pragma unroll
            for (int r = 0; r < 8; ++r) {
                int row = wm + (mi << 4) + (khalf << 3) + r;   // p
                int col = wn + (ni << 4) + lrow;               // n
                Sl[base + (size_t)row * DST + col] = acc[mi][ni][r];
            }
}

// =====================================================================
// inter-chunk recurrence:  P_prev[c] = P ;  P = exp(Acum_last[c]) * P + S_local[c]
// =====================================================================
__global__ __launch_bounds__(256) void chunk_recur(
    const float* __restrict__ Sl, const float* __restrict__ acum,
    float* __restrict__ Pp)
{
    const int id = blockIdx.x;            // b*32+h
    const int b = id >> 5, h = id & 31;
    const int tid = threadIdx.x;
    float P[64];
    #pragma unroll
    for (int i = 0; i < 64; ++i) P[i] = 0.f;
    for (int c = 0; c < NCH; ++c) {
        size_t base = (size_t)(((b * NCH) + c) * NH + h) * (HD * DST);
        float dA = __expf(acum[(((b * NH) + h) * NCH + c) * CHL + (CHL - 1)]);
        #pragma unroll
        for (int i = 0; i < 64; ++i) {
            size_t e = base + (size_t)i * 256 + tid;
            Pp[e] = P[i];
            P[i] = dA * P[i] + Sl[e];
        }
    }
}

// =====================================================================
// Fused Y via WMMA.  Per (b,c,h), l-tile of 128 x p=128:
//  phase1 (K = s, causal):  As[l][s] = G[l,s]*exp(A[l]-A[s])*dt[s] (s<=l)
//                           Bs[p][s] = x[s,p]            -> Y_diag
//  phase2 (K = n = 128):    As[l][n] = C[l,n]*exp(A[l])
//                           Bs[p][n] = P[p,n]            -> + Y_off
//  epilogue adds D[h]*x[l,p] and writes y.
// =====================================================================
__global__ __launch_bounds__(256) void ychunk_wmma(
    const float* __restrict__ xbc, const float* __restrict__ dtb,
    const float* __restrict__ acum, const float* __restrict__ G,
    const float* __restrict__ Pp, const float* __restrict__ Dp,
    float* __restrict__ yb)
{
    const int id = blockIdx.y;
    const int b = id >> 8, rem = id & 255, c = rem >> 5, h = rem & 31;
    const int tileM = blockIdx.x << 7;     // l offset within chunk (0 or 128)
    const size_t t0 = (size_t)b * SEQL + c * CHL;
    const int hbase = h * HD;

    __shared__ __align__(16) float As[128][34];
    __shared__ __align__(16) float Bs[128][34];
    __shared__ float sA[CHL];
    __shared__ float sDt[CHL];

    const int tid   = threadIdx.x;
    const int wave  = tid >> 5;
    const int lane  = tid & 31;
    const int wm    = (wave >> 1) << 5;
    const int wn    = (wave & 1) << 6;
    const int lrow  = lane & 15;
    const int khalf = lane >> 4;

    sA[tid]  = acum[(((b * NH) + h) * NCH + c) * CHL + tid];
    sDt[tid] = dtb[(t0 + tid) * NH + h];
    __syncthreads();

    const float* Gm = G + (size_t)((b * NCH) + c) * (CHL * CHL);
    v8f acc[2][4] = {};

    // ---- phase 1: causal diag part; s only needs to reach tileM+128 ----
    const int kend = tileM + 128;
    for (int k0 = 0; k0 < kend; k0 += 32) {
        #pragma unroll
        for (int i = 0; i < 4; ++i) {
            int idx = tid + (i << 8);
            // A: masked, decayed G   (rows s-contiguous -> coalesced v4 read)
            int r  = idx >> 3;
            int kq = (idx & 7) << 2;
            int l  = tileM + r;
            v4f gv = *(const v4f*)(Gm + (size_t)l * CHL + k0 + kq);
            float al = sA[l];
            #pragma unroll
            for (int q = 0; q < 4; ++q) {
                int s = k0 + kq + q;
                As[r][kq + q] = (s <= l) ? gv[q] * __expf(al - sA[s]) * sDt[s] : 0.f;
            }
            // B: transposed x tile (coalesced row reads, LDS scatter)
            int p  = idx & 127;
            int s4 = ((idx >> 7) & 7) << 2;
            #pragma unroll
            for (int q = 0; q < 4; ++q) {
                int s = k0 + s4 + q;
                Bs[p][s4 + q] = xbc[(t0 + s) * CONVD + hbase + p];
            }
        }
        __syncthreads();
        WMMA_TILE_STEP(As, Bs)
        __syncthreads();
    }

    // ---- phase 2: Y_off accumulated into same accumulators ----
    const float* Pm = Pp + (size_t)(((b * NCH) + c) * NH + h) * (HD * DST);
    for (int k0 = 0; k0 < DST; k0 += 32) {
        #pragma unroll
        for (int i = 0; i < 4; ++i) {
            int idx = tid + (i << 8);
            int r  = idx >> 3;
            int kq = (idx & 7) << 2;
            int l  = tileM + r;
            v4f cv = *(const v4f*)(xbc + (t0 + l) * CONVD + DINNER + DST + k0 + kq);
            float el = __expf(sA[l]);
            As[r][kq + 0] = cv[0] * el; As[r][kq + 1] = cv[1] * el;
            As[r][kq + 2] = cv[2] * el; As[r][kq + 3] = cv[3] * el;
            v4f pv = *(const v4f*)(Pm + (size_t)r * DST + k0 + kq);   // r = p row
            Bs[r][kq + 0] = pv[0]; Bs[r][kq + 1] = pv[1];
            Bs[r][kq + 2] = pv[2]; Bs[r][kq + 3] = pv[3];
        }
        __syncthreads();
        WMMA_TILE_STEP(As, Bs)
        __syncthreads();
    }

    // ---- epilogue: + D[h]*x, write y ----
    const float Dh = Dp[h];
    #pragma unroll
    for (int mi = 0; mi < 2; ++mi)
        #pragma unroll
        for (int ni = 0; ni < 4; ++ni)
            #pragma unroll
            for (int r = 0; r < 8; ++r) {
                int row = wm + (mi << 4) + (khalf << 3) + r;   // l local
                int col = wn + (ni << 4) + lrow;               // p
                int l = tileM + row;
                float xr = xbc[(t0 + l) * CONVD + hbase + col];
                yb[(t0 + l) * DINNER + hbase + col] = acc[mi][ni][r] + Dh * xr;
            }
}

// =====================================================================
// y = y * silu(z);  y = y * rsqrt(mean(y^2)+eps) * norm_w   (per token)
// =====================================================================
__global__ __launch_bounds__(256) void gate_rmsnorm(
    const float* __restrict__ zx, const float* __restrict__ normW,
    float* __restrict__ yb)
{
    const int t = blockIdx.x;
    const int tid = threadIdx.x;
    float vals[16];
    float ss = 0.f;
    #pragma unroll
    for (int j = 0; j < 16; ++j) {
        int i = tid + j * 256;
        float yv = yb[(size_t)t * DINNER + i];
        float zv = zx[(size_t)t * DIP + i];
        yv *= zv / (1.f + __expf(-zv));
        vals[j] = yv;
        ss += yv * yv;
    }
    __shared__ float red[256];
    red[tid] = ss;
    __syncthreads();
    for (int o = 128; o > 0; o >>= 1) {
        if (tid < o) red[tid] += red[tid + o];
        __syncthreads();
    }
    float scale = rsqrtf(red[0] / (float)DINNER + 1e-5f);
    #pragma unroll
    for (int j = 0; j < 16; ++j) {
        int i = tid + j * 256;
        yb[(size_t)t * DINNER + i] = vals[j] * scale * normW[i];
    }
}

// =====================================================================
extern "C" void kernel_launch(void* const* d_in, const int* in_sizes, int n_in,
                              void* d_out, int out_size, void* d_ws, size_t ws_size,
                              hipStream_t stream) {
    (void)in_sizes; (void)n_in; (void)out_size; (void)ws_size;
    const float* x_in   = (const float*)d_in[0];
    const float* inW    = (const float*)d_in[1];
    const float* inB    = (const float*)d_in[2];
    const float* outW   = (const float*)d_in[3];
    const float* outB   = (const float*)d_in[4];
    const float* convW  = (const float*)d_in[5];
    const float* convB  = (const float*)d_in[6];
    const float* dtBias = (const float*)d_in[7];
    const float* Alog   = (const float*)d_in[8];
    const float* Dp     = (const float*)d_in[9];
    const float* normW  = (const float*)d_in[10];
    float* out = (float*)d_out;
    float* ws  = (float*)d_ws;

    size_t o = 0;
    float* zx   = ws + o;  o += (size_t)TOK * DIP;                 // in_proj out
    float* xbc  = ws + o;  o += (size_t)TOK * CONVD;               // conv+silu out
    float* dtb  = ws + o;  o += (size_t)TOK * NH;                  // softplus dt
    float* acum = ws + o;  o += (size_t)2 * NH * NCH * CHL;        // A cumsum
    float* G    = ws + o;  o += (size_t)2 * NCH * CHL * CHL;       // C@B^T per chunk
    float* Sl   = ws + o;  o += (size_t)2 * NCH * NH * HD * DST;   // local states
    float* Pp   = ws + o;  o += (size_t)2 * NCH * NH * HD * DST;   // prev (carried) states
    float* yb   = ws + o;  o += (size_t)TOK * DINNER;              // y

    // 1) in_proj GEMM: zx = inputs @ inW^T + inB   (4096 x 8480, K=2048)
    {
        dim3 g((DIP + 127) / 128, TOK / 128, 1);
        gemm_nt_wmma<<<g, 256, 0, stream>>>(x_in, 0LL, DIMK, inW, 0LL, DIMK, inB,
                                            zx, 0LL, DIP, TOK, DIP, DIMK);
    }
    // 2) causal conv1d + SiLU + softplus(dt)
    conv_silu_dt<<<TOK, 256, 0, stream>>>(zx, convW, convB, dtBias, xbc, dtb);
    // 3) per-chunk cumsum of A*dt
    cumsum_kernel<<<2 * NH * NCH, 256, 0, stream>>>(dtb, Alog, acum);
    // 4) G = Cc @ Bc^T per (b,c): batched NT GEMM, 256x256, K=128
    {
        dim3 g(2, 2, 2 * NCH);
        gemm_nt_wmma<<<g, 256, 0, stream>>>(
            xbc + DINNER + DST, (long long)CHL * CONVD, CONVD,
            xbc + DINNER,       (long long)CHL * CONVD, CONVD,
            nullptr, G, (long long)CHL * CHL, CHL, CHL, CHL, DST);
    }
    // 5) local chunk states (WMMA)
    states_wmma<<<2 * NCH * NH, 256, 0, stream>>>(xbc, dtb, acum, Sl);
    // 6) inter-chunk recurrence
    chunk_recur<<<2 * NH, 256, 0, stream>>>(Sl, acum, Pp);
    // 7) Y = Y_diag + Y_off + D*x (WMMA, fused)
    {
        dim3 g(2, 2 * NCH * NH, 1);
        ychunk_wmma<<<g, 256, 0, stream>>>(xbc, dtb, acum, G, Pp, Dp, yb);
    }
    // 8) gate with silu(z) + RMSNorm
    gate_rmsnorm<<<TOK, 256, 0, stream>>>(zx, normW, yb);
    // 9) out_proj GEMM: out = yb @ outW^T + outB  (4096 x 2048, K=4096)
    {
        dim3 g(DIMK / 128, TOK / 128, 1);
        gemm_nt_wmma<<<g, 256, 0, stream>>>(yb, 0LL, DINNER, outW, 0LL, DINNER, outB,
                                            out, 0LL, DIMK, TOK, DIMK, DINNER);
    }
}